// GcnModel_7164005449946
// MI455X (gfx1250) — compile-verified
//
#include <hip/hip_runtime.h>
#include <hip/hip_bf16.h>

#define N_NODES   20000
#define N_EDGES   320000
#define HID       256
#define N_GRAPHS  512
#define N_LAYERS  3

typedef float v2f __attribute__((ext_vector_type(2)));
typedef float v8f __attribute__((ext_vector_type(8)));
typedef int   v4i __attribute__((ext_vector_type(4)));

#if defined(__has_builtin)
#if __has_builtin(__builtin_amdgcn_global_load_async_to_lds_b128) && \
    __has_builtin(__builtin_amdgcn_s_wait_asynccnt)
#define HAVE_ASYNC_COPY 1
#endif
#endif

static __device__ __forceinline__ void atomic_add_f32(float* p, float v) {
    __hip_atomic_fetch_add(p, v, __ATOMIC_RELAXED, __HIP_MEMORY_SCOPE_AGENT);
}

// 16B async (or sync fallback) copy global -> LDS
static __device__ __forceinline__ void copy16_g2l(const float* gsrc, float* ldst) {
#if HAVE_ASYNC_COPY
    __builtin_amdgcn_global_load_async_to_lds_b128(
        (v4i*)gsrc, (v4i*)ldst, /*offset=*/0, /*cpol=*/0);
#else
    *(float4*)ldst = *(const float4*)gsrc;
#endif
}

static __device__ __forceinline__ void wait_async_copies() {
#if HAVE_ASYNC_COPY
    __builtin_amdgcn_s_wait_asynccnt(0);
#endif
}

// ---------------- small elementwise kernels ----------------

__global__ void k_init_deg(float* deg) {
    int i = blockIdx.x * blockDim.x + threadIdx.x;
    if (i < N_NODES) deg[i] = 1.0f;   // self-loop
}

__global__ void k_count_deg(const int* __restrict__ dst, float* deg) {
    int e = blockIdx.x * blockDim.x + threadIdx.x;
    if (e < N_EDGES) atomic_add_f32(&deg[dst[e]], 1.0f);
}

__global__ void k_rsqrt(float* dis) {
    int i = blockIdx.x * blockDim.x + threadIdx.x;
    if (i < N_NODES) dis[i] = rsqrtf(dis[i]);
}

// h[n, :] = embd_w[x[n], :]
__global__ void k_embed(const int* __restrict__ x, const float* __restrict__ embd,
                        float* __restrict__ h) {
    int tid = blockIdx.x * blockDim.x + threadIdx.x;
    if (tid >= N_NODES * (HID / 4)) return;
    int n = tid >> 6;
    int q = tid & 63;
    const float4* E4 = (const float4*)embd;
    ((float4*)h)[tid] = E4[(size_t)x[n] * (HID / 4) + q];
}

// agg[n,c] = hw[n,c] * dis[n]^2 + bias[c]
__global__ void k_init_agg(const float* __restrict__ hw, const float* __restrict__ dis,
                           const float* __restrict__ bias, float* __restrict__ agg) {
    int tid = blockIdx.x * blockDim.x + threadIdx.x;
    if (tid >= N_NODES * (HID / 4)) return;
    int n = tid >> 6;
    int q = tid & 63;
    float sn = dis[n] * dis[n];
    float4 v = ((const float4*)hw)[tid];
    float4 b = ((const float4*)bias)[q];
    float4 o;
    o.x = v.x * sn + b.x; o.y = v.y * sn + b.y;
    o.z = v.z * sn + b.z; o.w = v.w * sn + b.w;
    ((float4*)agg)[tid] = o;
}

// agg[dst[e], c] += hw[src[e], c] * dis[src]*dis[dst]
__global__ void k_edge_scatter(const int* __restrict__ src, const int* __restrict__ dst,
                               const float* __restrict__ dis, const float* __restrict__ hw,
                               float* __restrict__ agg) {
    int tid = blockIdx.x * blockDim.x + threadIdx.x;
    if (tid >= N_EDGES * (HID / 4)) return;
    int e = tid >> 6;
    int q = tid & 63;
    int s = src[e], d = dst[e];
    float w = dis[s] * dis[d];
    float4 v = ((const float4*)hw)[(size_t)s * (HID / 4) + q];
    float* out = agg + (size_t)d * HID + q * 4;
    atomic_add_f32(out + 0, v.x * w);
    atomic_add_f32(out + 1, v.y * w);
    atomic_add_f32(out + 2, v.z * w);
    atomic_add_f32(out + 3, v.w * w);
}

// s[n] = dot(z[n,:], w2) + b2
__global__ void k_gemv_head(const float* __restrict__ z, const float* __restrict__ w2,
                            const float* __restrict__ b2, float* __restrict__ s) {
    int n = blockIdx.x * blockDim.x + threadIdx.x;
    if (n >= N_NODES) return;
    const float4* zr = (const float4*)(z + (size_t)n * HID);
    const float4* w4 = (const float4*)w2;
    float acc = 0.0f;
#pragma unroll 8
    for (int i = 0; i < HID / 4; ++i) {
        float4 a = zr[i], b = w4[i];
        acc += a.x * b.x + a.y * b.y + a.z * b.z + a.w * b.w;
    }
    s[n] = acc + b2[0];
}

__global__ void k_zero_out(float* out) {
    int i = blockIdx.x * blockDim.x + threadIdx.x;
    if (i < N_GRAPHS) out[i] = 0.0f;
}

__global__ void k_pool(const int* __restrict__ batch, const float* __restrict__ s,
                       float* __restrict__ out) {
    int n = blockIdx.x * blockDim.x + threadIdx.x;
    if (n < N_NODES) atomic_add_f32(&out[batch[n]], s[n]);
}

// ---------------- weight repack into WMMA fragment order ----------------
// Bp[kg*1024 + n*4 + kl] = B[(4*kg + kl)*256 + n]
// so a lane's 16x4 B fragment (n = col, kl = 2*hiHalf + {0,1}) is one aligned b64.
__global__ void k_pack_B(const float* __restrict__ B, float* __restrict__ Bp) {
    int idx = blockIdx.x * blockDim.x + threadIdx.x;
    if (idx >= HID * HID) return;
    int kg = idx >> 10;
    int r  = idx & 1023;
    int n  = r >> 2;
    int kl = r & 3;
    Bp[idx] = B[(size_t)(4 * kg + kl) * HID + n];
}

// ---------------- WMMA fp32 GEMM ----------------
// C[M,256] = act(A)[M,256] @ B[256,256] (+bias) (+relu_out)
// Block = 256 threads = 8 waves; wave w owns rows [blk*128+16w, +16) x 256 cols
// (16 accumulator tiles of v_wmma_f32_16x16x4_f32).
// Packed B is staged through double-buffered 32KB LDS chunks (async if available).

#define KROWS_CHUNK 32                       // K rows per chunk
#define CH_FLOATS   (KROWS_CHUNK * HID)      // 8192 floats = 32KB
#define N_CHUNKS    (HID / KROWS_CHUNK)      // 8

template <int RELU_IN, int RELU_OUT, int HAS_BIAS>
__global__ __launch_bounds__(256)
void k_gemm_wmma(const float* __restrict__ A, const float* __restrict__ Bp,
                 const float* __restrict__ bias, float* __restrict__ C, int M) {
    __shared__ float sB[2][CH_FLOATS];       // 64 KB total

    const int tid    = threadIdx.x;
    const int wave   = tid >> 5;
    const int lane   = tid & 31;
    const int laneLo = lane & 15;
    const int hiHalf = lane >> 4;
    const int kOff   = hiHalf * 2;
    const int mOff   = hiHalf * 8;

    const int rowBase = blockIdx.x * 128 + wave * 16;
    int mA = rowBase + laneLo;
    if (mA > M - 1) mA = M - 1;              // clamp loads; stores guarded

    v8f acc[16];
#pragma unroll
    for (int nt = 0; nt < 16; ++nt) acc[nt] = (v8f)(0.0f);

    const float* Arow = A + (size_t)mA * HID;

    // stage chunk 0
#pragma unroll
    for (int i = 0; i < CH_FLOATS / 4 / 256; ++i) {
        int u = (i * 256 + tid) * 4;
        copy16_g2l(Bp + u, &sB[0][u]);
    }
    wait_async_copies();
    __syncthreads();

    for (int c = 0; c < N_CHUNKS; ++c) {
        const int cur = c & 1;
        if (c + 1 < N_CHUNKS) {
            const float* gch = Bp + (size_t)(c + 1) * CH_FLOATS;
#pragma unroll
            for (int i = 0; i < CH_FLOATS / 4 / 256; ++i) {
                int u = (i * 256 + tid) * 4;
                copy16_g2l(gch + u, &sB[cur ^ 1][u]);
            }
        }

        const float* Achunk = Arow + c * KROWS_CHUNK;
#pragma unroll
        for (int kg = 0; kg < KROWS_CHUNK / 4; ++kg) {
            v2f a = *(const v2f*)(Achunk + kg * 4 + kOff);
            if (RELU_IN) { a.x = fmaxf(a.x, 0.0f); a.y = fmaxf(a.y, 0.0f); }

            const float* sfrag = &sB[cur][kg * 1024 + laneLo * 4 + kOff];
#pragma unroll
            for (int nt = 0; nt < 16; ++nt) {
                v2f b = *(const v2f*)(sfrag + nt * 64);   // single ds_load_b64
                acc[nt] = __builtin_amdgcn_wmma_f32_16x16x4_f32(
                    false, a, false, b, (short)0, acc[nt], false, false);
            }
        }

        wait_async_copies();
        __syncthreads();
    }

    // store: VGPR r of tile nt -> C[rowBase + r + mOff, nt*16 + laneLo]
#pragma unroll
    for (int nt = 0; nt < 16; ++nt) {
        const int n = nt * 16 + laneLo;
        const float bv = HAS_BIAS ? bias[n] : 0.0f;
#pragma unroll
        for (int r = 0; r < 8; ++r) {
            int m = rowBase + r + mOff;
            if (m < M) {
                float v = acc[nt][r] + bv;
                if (RELU_OUT) v = fmaxf(v, 0.0f);
                C[(size_t)m * HID + n] = v;
            }
        }
    }
}

// ---------------- launch ----------------

extern "C" void kernel_launch(void* const* d_in, const int* in_sizes, int n_in,
                              void* d_out, int out_size, void* d_ws, size_t ws_size,
                              hipStream_t stream) {
    const int*   x      = (const int*)  d_in[0];
    const int*   edge   = (const int*)  d_in[1];   // [2, N_EDGES]
    const int*   batch  = (const int*)  d_in[2];
    const float* embd_w = (const float*)d_in[3];
    const float* conv_w = (const float*)d_in[4];   // [3,256,256]
    const float* conv_b = (const float*)d_in[5];   // [3,256]
    const float* mlp_w1 = (const float*)d_in[6];
    const float* mlp_b1 = (const float*)d_in[7];
    const float* mlp_w2 = (const float*)d_in[8];
    const float* mlp_b2 = (const float*)d_in[9];
    float*       out    = (float*)d_out;

    const int* src = edge;
    const int* dst = edge + N_EDGES;

    // workspace carve-up (blocks packed back to back, no padding)
    char* ws = (char*)d_ws;
    size_t off = 0;
    float* bufA = (float*)(ws + off); off += (size_t)N_NODES * HID * 4;   // h / h_next
    float* bufB = (float*)(ws + off); off += (size_t)N_NODES * HID * 4;   // hw / z
    float* bufP = (float*)(ws + off); off += (size_t)HID * HID * 4;       // packed weights
    float* dis  = (float*)(ws + off); off += (size_t)N_NODES * 4;
    float* svec = (float*)(ws + off);

    const int T = 256;
    dim3 blk(T);

    // degree -> D^{-1/2}
    k_init_deg <<<(N_NODES + T - 1) / T, blk, 0, stream>>>(dis);
    k_count_deg<<<(N_EDGES + T - 1) / T, blk, 0, stream>>>(dst, dis);
    k_rsqrt    <<<(N_NODES + T - 1) / T, blk, 0, stream>>>(dis);

    // embedding lookup -> bufA
    k_embed<<<(N_NODES * (HID / 4) + T - 1) / T, blk, 0, stream>>>(x, embd_w, bufA);

    const int gemm_blocks = (N_NODES + 127) / 128;
    const int pack_blocks = (HID * HID + T - 1) / T;
    const int ew_blocks   = (N_NODES * (HID / 4) + T - 1) / T;
    const int sc_blocks   = (N_EDGES * (HID / 4) + T - 1) / T;

    for (int i = 0; i < N_LAYERS; ++i) {
        k_pack_B<<<pack_blocks, blk, 0, stream>>>(conv_w + (size_t)i * HID * HID, bufP);
        if (i == 0)
            k_gemm_wmma<0, 0, 0><<<gemm_blocks, blk, 0, stream>>>(bufA, bufP, nullptr, bufB, N_NODES);
        else
            k_gemm_wmma<1, 0, 0><<<gemm_blocks, blk, 0, stream>>>(bufA, bufP, nullptr, bufB, N_NODES);
        // h_next = hw * self_norm + bias  (into bufA; old A dead)
        k_init_agg<<<ew_blocks, blk, 0, stream>>>(bufB, dis, conv_b + (size_t)i * HID, bufA);
        // h_next[dst] += hw[src] * enorm
        k_edge_scatter<<<sc_blocks, blk, 0, stream>>>(src, dst, dis, bufB, bufA);
    }

    // z = relu(h3 @ mlp_w1 + b1) -> bufB
    k_pack_B<<<pack_blocks, blk, 0, stream>>>(mlp_w1, bufP);
    k_gemm_wmma<0, 1, 1><<<gemm_blocks, blk, 0, stream>>>(bufA, bufP, mlp_b1, bufB, N_NODES);

    // s[n] = z[n] . w2 + b2
    k_gemv_head<<<(N_NODES + T - 1) / T, blk, 0, stream>>>(bufB, mlp_w2, mlp_b2, svec);

    // per-graph pooling
    k_zero_out<<<(N_GRAPHS + T - 1) / T, blk, 0, stream>>>(out);
    k_pool    <<<(N_NODES + T - 1) / T, blk, 0, stream>>>(batch, svec, out);
}